// GraphAttention_47132971106391
// MI455X (gfx1250) — compile-verified
//
#include <hip/hip_runtime.h>
#include <stdint.h>

// ---------------------------------------------------------------------------
// GAT layer for MI455X (gfx1250, wave32).
//   Phase 1: h = X * W   -> v_wmma_f32_16x16x4_f32, W staged via TDM
//            (tensor_load_to_lds) then swizzled into fragment-native LDS
//            layout so each B fragment pair is ONE ds_load_b128.
//   Phase 2: score_dst/src = h . a_dst / a_src (wave-per-node dot)
//   Phase 3: per-edge exp(clip(leaky_relu(...))) + segment-sum (f32 atomics)
//   Phase 4: out[dst] += h[src] * attn  (wave-per-edge, float4/lane)
// ---------------------------------------------------------------------------

typedef __attribute__((ext_vector_type(2))) float v2f;
typedef __attribute__((ext_vector_type(4))) float v4f;
typedef __attribute__((ext_vector_type(8))) float v8f;
typedef __attribute__((ext_vector_type(4))) unsigned int v4u;
typedef __attribute__((ext_vector_type(4))) int v4i;
typedef __attribute__((ext_vector_type(8))) int v8i;

#define FEAT 128     // F_IN == UNITS == 128
#define KSTEPS 32    // 128 / 4
#define SWZ_ENTRIES (KSTEPS * 8 * 32)   // (kstep, nt, lane) float2 entries

#if defined(__HIP_DEVICE_COMPILE__) && defined(__gfx1250__) && \
    __has_builtin(__builtin_amdgcn_tensor_load_to_lds) && \
    __has_builtin(__builtin_amdgcn_s_wait_tensorcnt)
#define USE_TDM 1
#else
#define USE_TDM 0
#endif

// ---------------------------------------------------------------------------
// Kernel 1: H = X (n x 128) * W (128 x 128), fp32 WMMA.
// Block = 128 threads (4 waves). Each wave computes a 16-row x 128-col strip.
// ---------------------------------------------------------------------------
__global__ __launch_bounds__(128) void gat_gemm_kernel(
    const float* __restrict__ X, const float* __restrict__ W,
    float* __restrict__ H, int n_nodes)
{
    // Fragment-native swizzled W: [kstep][ntpair][lane][part] as float2;
    // a lane's two consecutive entries (part=0,1) are read as one b128.
    __shared__ __align__(16) v2f swz[SWZ_ENTRIES];

#if USE_TDM
    __shared__ __align__(16) float raww[FEAT * FEAT];   // raw W tile via TDM

    if (threadIdx.x < 32) {
        // ---- Tensor DMA descriptor (D#) for a 2D 128x128 fp32 tile ----
        const unsigned lds_addr = (unsigned)(uintptr_t)(&raww[0]);
        const unsigned long long ga = (unsigned long long)(uintptr_t)W;

        v4u g0 = {0u, 0u, 0u, 0u};
        g0[0] = 1u;                                         // count=1, user mode
        g0[1] = lds_addr;                                   // lds_addr [63:32]
        g0[2] = (unsigned)(ga & 0xFFFFFFFFu);               // global_addr lo
        g0[3] = (unsigned)((ga >> 32) & 0x01FFFFFFu)        // global_addr hi
              | 0x80000000u;                                // type=2 ("image")

        v8i g1 = {0, 0, 0, 0, 0, 0, 0, 0};
        g1[0] = (int)(2u << 16);        // data_size=2 (4 bytes), no multicast
        g1[1] = (int)(128u << 16);      // tensor_dim0 lo16 in [63:48]
        g1[2] = (int)(128u << 16);      // tensor_dim0 hi16=0, tensor_dim1 lo16
        g1[3] = (int)(128u << 16);      // tensor_dim1 hi16=0, tile_dim0=128
        g1[4] = (int)(128u);            // tile_dim1=128, tile_dim2=0
        g1[5] = (int)(128u);            // tensor_dim0_stride lo32 = 128
        g1[6] = 0;                      // stride hi16 / dim1_stride lo16
        g1[7] = 0;

        v4i z4 = {0, 0, 0, 0};
#if __clang_major__ >= 23
        v8i z8 = {0, 0, 0, 0, 0, 0, 0, 0};
        __builtin_amdgcn_tensor_load_to_lds(g0, g1, z4, z4, z8, 0);
#else
        __builtin_amdgcn_tensor_load_to_lds(g0, g1, z4, z4, 0);
#endif
        __builtin_amdgcn_s_wait_tensorcnt(0);
    }
    __syncthreads();
    const float* __restrict__ SRC = &raww[0];
#else
    const float* __restrict__ SRC = W;
#endif

    // Swizzle SRC (row-major 128x128) into fragment layout.
    for (int e = threadIdx.x; e < SWZ_ENTRIES; e += blockDim.x) {
        const int kstep = e >> 8;
        const int ntp   = (e >> 6) & 3;
        const int ln    = (e >> 1) & 31;
        const int part  = e & 1;
        const int nt    = ntp * 2 + part;
        const int k     = kstep * 4 + ((ln >> 4) << 1);
        const int col   = nt * 16 + (ln & 15);
        v2f t;
        t.x = SRC[k * FEAT + col];
        t.y = SRC[(k + 1) * FEAT + col];
        swz[e] = t;
    }
    __syncthreads();

    const int wave  = threadIdx.x >> 5;
    const int lane  = threadIdx.x & 31;
    const int m     = lane & 15;
    const int khalf = (lane >> 4) << 1;     // 0 for lanes 0-15, 2 for 16-31

    const int row_base = blockIdx.x * 64 + wave * 16;

    int arow = row_base + m;                 // clamped; writes are guarded
    if (arow >= n_nodes) arow = n_nodes - 1;
    const float* __restrict__ ap = X + (long)arow * FEAT;

    const v4f* __restrict__ swz4 = reinterpret_cast<const v4f*>(&swz[0]);

    v8f acc[8];
    #pragma unroll
    for (int nt = 0; nt < 8; ++nt) acc[nt] = v8f{};

    #pragma unroll 4
    for (int kstep = 0; kstep < KSTEPS; ++kstep) {
        // A frag 16x4: lanes 0-15 hold K=k0,k0+1; lanes 16-31 K=k0+2,k0+3
        v2f a = *reinterpret_cast<const v2f*>(ap + kstep * 4 + khalf);

        #pragma unroll
        for (int ntp = 0; ntp < 4; ++ntp) {
            // One b128: two B fragments (n-tiles 2*ntp and 2*ntp+1)
            v4f bb = swz4[(kstep * 4 + ntp) * 32 + lane];
            acc[2 * ntp] = __builtin_amdgcn_wmma_f32_16x16x4_f32(
                false, a, false, bb.xy, (short)0, acc[2 * ntp], false, false);
            acc[2 * ntp + 1] = __builtin_amdgcn_wmma_f32_16x16x4_f32(
                false, a, false, bb.zw, (short)0, acc[2 * ntp + 1], false, false);
        }
    }

    // C/D layout: vgpr j -> row (j + 8*(lane>=16)), col = lane&15 per N-tile.
    const int row_off = (lane >> 4) << 3;
    #pragma unroll
    for (int nt = 0; nt < 8; ++nt) {
        #pragma unroll
        for (int j = 0; j < 8; ++j) {
            int r = row_base + row_off + j;
            if (r < n_nodes) H[(long)r * FEAT + nt * 16 + m] = acc[nt][j];
        }
    }
}

// ---------------------------------------------------------------------------
// Kernel 2: per-node attention scores (one wave per node, float4 per lane).
// ---------------------------------------------------------------------------
__global__ __launch_bounds__(256) void gat_scores_kernel(
    const float* __restrict__ H, const float* __restrict__ katt,
    float* __restrict__ sdst, float* __restrict__ ssrc, int n_nodes)
{
    const int gw   = (blockIdx.x * blockDim.x + threadIdx.x) >> 5;
    const int lane = threadIdx.x & 31;
    if (gw >= n_nodes) return;

    const float4 h4 = reinterpret_cast<const float4*>(H + (long)gw * FEAT)[lane];
    const float4 ad = reinterpret_cast<const float4*>(katt)[lane];          // a_dst
    const float4 as = reinterpret_cast<const float4*>(katt + FEAT)[lane];   // a_src

    float d = h4.x * ad.x + h4.y * ad.y + h4.z * ad.z + h4.w * ad.w;
    float s = h4.x * as.x + h4.y * as.y + h4.z * as.z + h4.w * as.w;

    #pragma unroll
    for (int off = 16; off > 0; off >>= 1) {
        d += __shfl_xor(d, off, 32);
        s += __shfl_xor(s, off, 32);
    }
    if (lane == 0) { sdst[gw] = d; ssrc[gw] = s; }
}

// ---------------------------------------------------------------------------
// Kernel 3: zero-init out and scores_sum.
// ---------------------------------------------------------------------------
__global__ void gat_zero_kernel(float* __restrict__ out, long n_out,
                                float* __restrict__ ssum, int n_sum)
{
    const long i = (long)blockIdx.x * blockDim.x + threadIdx.x;
    const long stride = (long)gridDim.x * blockDim.x;
    for (long j = i; j < n_out; j += stride) out[j] = 0.0f;
    for (long j = i; j < n_sum; j += stride) ssum[j] = 0.0f;
}

// ---------------------------------------------------------------------------
// Kernel 4: per-edge softmax numerator + denominator segment-sum.
// ---------------------------------------------------------------------------
__global__ __launch_bounds__(256) void gat_edge_score_kernel(
    const int* __restrict__ edges, const float* __restrict__ sdst,
    const float* __restrict__ ssrc, float* __restrict__ scores,
    float* __restrict__ ssum, int n_edges)
{
    const int e = blockIdx.x * blockDim.x + threadIdx.x;
    if (e >= n_edges) return;
    const int dst = edges[2 * e + 0];
    const int src = edges[2 * e + 1];
    float x = sdst[dst] + ssrc[src];
    x = (x > 0.0f) ? x : 0.2f * x;              // leaky_relu(0.2)
    x = fminf(fmaxf(x, -2.0f), 2.0f);           // clip
    const float sc = __expf(x);
    scores[e] = sc;
    atomicAdd(&ssum[dst], sc);
}

// ---------------------------------------------------------------------------
// Kernel 5: weighted scatter, one wave per edge, float4 per lane.
// ---------------------------------------------------------------------------
__global__ __launch_bounds__(256) void gat_scatter_kernel(
    const int* __restrict__ edges, const float* __restrict__ H,
    const float* __restrict__ scores, const float* __restrict__ ssum,
    float* __restrict__ out, int n_edges)
{
    const int gw   = (blockIdx.x * blockDim.x + threadIdx.x) >> 5;
    const int lane = threadIdx.x & 31;
    if (gw >= n_edges) return;

    const int dst = edges[2 * gw + 0];
    const int src = edges[2 * gw + 1];
    const float w = scores[gw] / ssum[dst];

    const float4 h4 = reinterpret_cast<const float4*>(H + (long)src * FEAT)[lane];
    float* op = out + (long)dst * FEAT + lane * 4;
    atomicAdd(op + 0, h4.x * w);
    atomicAdd(op + 1, h4.y * w);
    atomicAdd(op + 2, h4.z * w);
    atomicAdd(op + 3, h4.w * w);
}

// ---------------------------------------------------------------------------
// Launcher
// ---------------------------------------------------------------------------
extern "C" void kernel_launch(void* const* d_in, const int* in_sizes, int n_in,
                              void* d_out, int out_size, void* d_ws, size_t ws_size,
                              hipStream_t stream)
{
    const float* X    = (const float*)d_in[0];   // node_states (N x 128)
    const int*   E    = (const int*)d_in[1];     // edges (M x 2), [dst, src]
    const float* W    = (const float*)d_in[2];   // kernel (128 x 128)
    const float* katt = (const float*)d_in[3];   // kernel_attention (256 x 1)

    const int n_nodes = in_sizes[0] / FEAT;
    const int n_edges = in_sizes[1] / 2;

    float* ws     = (float*)d_ws;
    float* H      = ws;                                  // n_nodes * 128
    float* sdst   = H + (long)n_nodes * FEAT;            // n_nodes
    float* ssrc   = sdst + n_nodes;                      // n_nodes
    float* ssum   = ssrc + n_nodes;                      // n_nodes
    float* scores = ssum + n_nodes;                      // n_edges
    float* out    = (float*)d_out;                       // n_nodes * 128

    gat_gemm_kernel<<<(n_nodes + 63) / 64, 128, 0, stream>>>(X, W, H, n_nodes);
    gat_scores_kernel<<<(n_nodes + 7) / 8, 256, 0, stream>>>(H, katt, sdst, ssrc, n_nodes);
    gat_zero_kernel<<<1024, 256, 0, stream>>>(out, (long)n_nodes * FEAT, ssum, n_nodes);
    gat_edge_score_kernel<<<(n_edges + 255) / 256, 256, 0, stream>>>(
        E, sdst, ssrc, scores, ssum, n_edges);
    gat_scatter_kernel<<<(n_edges + 7) / 8, 256, 0, stream>>>(
        E, H, scores, ssum, out, n_edges);
}